// ENSELayer_970662609557
// MI455X (gfx1250) — compile-verified
//
#include <hip/hip_runtime.h>
#include <math.h>

typedef __attribute__((ext_vector_type(2))) float v2f;
typedef __attribute__((ext_vector_type(8))) float v8f;

// ---------------------------------------------------------------------------
// Kernel 1: fold w1 (1,3,3,3), w2 (1,1,3,3) and the pooling normalizations
// into a single 4x4 weighting Kw over the per-channel y4 pool:
//   s[b,c] = sum_{u,v} Kw[u,v] * (sum of x over the 14x14 block (u,v))
// ---------------------------------------------------------------------------
__global__ void keff_kernel(const float* __restrict__ w1,
                            const float* __restrict__ w2,
                            float* __restrict__ kw) {
    if (threadIdx.x != 0) return;
    // conv2 (pad 1, stride 3, k=3 on 2x2 input) only sees w2[1+p][1+q]
    float w2c[2][2];
    for (int p = 0; p < 2; ++p)
        for (int q = 0; q < 2; ++q)
            w2c[p][q] = w2[(1 + p) * 3 + (1 + q)];
    // coeff[m][u][v]: weight of t[m][u][v] in the final scalar
    float coeff[3][4][4];
    for (int m = 0; m < 3; ++m)
        for (int u = 0; u < 4; ++u)
            for (int v = 0; v < 4; ++v) {
                float c = 0.f;
                for (int p = 0; p < 2; ++p)
                    for (int q = 0; q < 2; ++q) {
                        int i = u - p, j = v - q;
                        if (i >= 0 && i < 3 && j >= 0 && j < 3)
                            c += w2c[p][q] * w1[m * 9 + i * 3 + j];
                    }
                coeff[m][u][v] = c;
            }
    // y1 = mean of all 16 y4 cells; y2(upsampled)[u][v] = mean of 2x2 y4 block
    float tot0 = 0.f;
    for (int u = 0; u < 4; ++u)
        for (int v = 0; v < 4; ++v) tot0 += coeff[0][u][v];
    float bs1[2][2] = {{0.f, 0.f}, {0.f, 0.f}};
    for (int u = 0; u < 4; ++u)
        for (int v = 0; v < 4; ++v) bs1[u >> 1][v >> 1] += coeff[1][u][v];
    for (int u = 0; u < 4; ++u)
        for (int v = 0; v < 4; ++v) {
            float k = coeff[2][u][v] + bs1[u >> 1][v >> 1] * 0.25f + tot0 * (1.f / 16.f);
            kw[u * 4 + v] = k * (1.f / 196.f);   // fold 14x14 mean
        }
}

// ---------------------------------------------------------------------------
// Kernel 2: per-(b,c) weighted plane reduction (the 205 MB read).
// One 256-thread block per plane; float4-coalesced loads; LDS tree reduce.
// ---------------------------------------------------------------------------
__global__ void __launch_bounds__(256) pool_kernel(const float* __restrict__ x,
                                                   const float* __restrict__ kw,
                                                   float* __restrict__ sbuf) {
    __shared__ float red[256];
    __shared__ float kws[16];
    const int tid = threadIdx.x;
    const int plane = blockIdx.x;                  // b*512 + c
    if (tid < 16) kws[tid] = kw[tid];
    __syncthreads();

    const float4* xp4 = (const float4*)(x + (size_t)plane * 3136);
    float acc = 0.f;
    for (int e4 = tid; e4 < 784; e4 += 256) {      // 3136/4 float4s
        float4 d = xp4[e4];
        int e = e4 * 4;
        int row = e / 56;
        int col = e - row * 56;                    // 56%4==0 -> same row for all 4
        int u4 = (row / 14) * 4;
        acc += d.x * kws[u4 + (col    ) / 14];
        acc += d.y * kws[u4 + (col + 1) / 14];
        acc += d.z * kws[u4 + (col + 2) / 14];
        acc += d.w * kws[u4 + (col + 3) / 14];
    }
    red[tid] = acc;
    __syncthreads();
    for (int s = 128; s > 0; s >>= 1) {
        if (tid < s) red[tid] += red[tid + s];
        __syncthreads();
    }
    if (tid == 0) sbuf[plane] = red[0];
}

// ---------------------------------------------------------------------------
// Kernel 3: sig = sigmoid( relu(S @ fc1^T) @ fc2^T ), S is 32x512.
// Single wave32, V_WMMA_F32_16X16X4_F32 (D = A(16x4) x B(4x16) + C).
// A-frag layout: lane<16 holds K=0,1 of row M=lane; lane>=16 holds K=2,3.
// D layout: vgpr g holds (M = g + 8*(lane>=16), N = lane%16).
// ---------------------------------------------------------------------------
__global__ void __launch_bounds__(32) fc_wmma_kernel(const float* __restrict__ s,
                                                     const float* __restrict__ fc1,
                                                     const float* __restrict__ fc2,
                                                     float* __restrict__ sig) {
    __shared__ float h[32 * 32];
    const int lane  = threadIdx.x;      // 0..31
    const int lm    = lane & 15;        // M (A) / N (B,D) sub-index
    const int khalf = (lane >> 4) * 2;  // K offset 0 or 2
    const int rbhi  = (lane >> 4) * 8;  // D row offset

    // ---- Stage 1: H(32x32) = relu( S(32x512) @ fc1^T ), K=512 ----
    for (int mi = 0; mi < 2; ++mi) {
        const float* arow = s + (size_t)(mi * 16 + lm) * 512;
        for (int ni = 0; ni < 2; ++ni) {
            const float* brow = fc1 + (size_t)(ni * 16 + lm) * 512;  // fc1[j, :]
            v8f acc = {};
            for (int k = 0; k < 512; k += 4) {
                v2f a, b;
                a.x = arow[k + khalf];
                a.y = arow[k + khalf + 1];
                b.x = brow[k + khalf];
                b.y = brow[k + khalf + 1];
                acc = __builtin_amdgcn_wmma_f32_16x16x4_f32(
                    false, a, false, b, (short)0, acc, false, false);
            }
            int rbase = mi * 16 + rbhi;
#pragma unroll
            for (int g = 0; g < 8; ++g)
                h[(rbase + g) * 32 + ni * 16 + lm] = fmaxf(acc[g], 0.f);
        }
    }
    __syncthreads();

    // ---- Stage 2: S2(32x512) = H @ fc2^T, K=32; sigmoid; store ----
    for (int mi = 0; mi < 2; ++mi) {
        for (int nc = 0; nc < 32; ++nc) {
            const float* brow = fc2 + (size_t)(nc * 16 + lm) * 32;   // fc2[c, :]
            v8f acc = {};
#pragma unroll
            for (int k = 0; k < 32; k += 4) {
                v2f a, b;
                a.x = h[(mi * 16 + lm) * 32 + k + khalf];
                a.y = h[(mi * 16 + lm) * 32 + k + khalf + 1];
                b.x = brow[k + khalf];
                b.y = brow[k + khalf + 1];
                acc = __builtin_amdgcn_wmma_f32_16x16x4_f32(
                    false, a, false, b, (short)0, acc, false, false);
            }
            int rbase = mi * 16 + rbhi;
#pragma unroll
            for (int g = 0; g < 8; ++g) {
                float v = acc[g];
                sig[(size_t)(rbase + g) * 512 + nc * 16 + lm] =
                    1.f / (1.f + __expf(-v));
            }
        }
    }
}

// ---------------------------------------------------------------------------
// Kernel 4: out = x * sig[b,c]  (the 411 MB stream), float4 vectorized.
// ---------------------------------------------------------------------------
__global__ void __launch_bounds__(256) scale_kernel(const float4* __restrict__ x,
                                                    const float* __restrict__ sig,
                                                    float4* __restrict__ out,
                                                    int n4) {
    int i = blockIdx.x * blockDim.x + threadIdx.x;
    if (i >= n4) return;
    int plane = i / 784;                           // 3136/4 float4 per plane
    float sc = sig[plane];
    float4 v = x[i];
    out[i] = make_float4(v.x * sc, v.y * sc, v.z * sc, v.w * sc);
}

// ---------------------------------------------------------------------------
extern "C" void kernel_launch(void* const* d_in, const int* in_sizes, int n_in,
                              void* d_out, int out_size, void* d_ws, size_t ws_size,
                              hipStream_t stream) {
    const float* x   = (const float*)d_in[0];   // 32*512*56*56
    const float* w1  = (const float*)d_in[1];   // 27
    const float* w2  = (const float*)d_in[2];   // 9
    const float* fc1 = (const float*)d_in[3];   // 32*512
    const float* fc2 = (const float*)d_in[4];   // 512*32
    float* out = (float*)d_out;

    char* ws = (char*)d_ws;
    float* kw   = (float*)(ws);                  // 16 floats
    float* sbuf = (float*)(ws + 128);            // 16384 floats
    float* sig  = (float*)(ws + 128 + 16384 * sizeof(float));  // 16384 floats

    const int planes = 32 * 512;                 // 16384
    const int n4 = 32 * 512 * 784;               // 12,845,056 float4s

    keff_kernel<<<1, 32, 0, stream>>>(w1, w2, kw);
    pool_kernel<<<planes, 256, 0, stream>>>(x, kw, sbuf);
    fc_wmma_kernel<<<1, 32, 0, stream>>>(sbuf, fc1, fc2, sig);
    scale_kernel<<<(n4 + 255) / 256, 256, 0, stream>>>((const float4*)x, sig,
                                                       (float4*)out, n4);
}